// HGNNModel_9740985828002
// MI455X (gfx1250) — compile-verified
//
#include <hip/hip_runtime.h>

#define DIM        128
#define LEAKY_K    0.2f
#define LN_EPS_K   1e-5f
#define TILE       256        // edges staged per block tile (== blockDim)

#if __has_builtin(__builtin_amdgcn_global_load_async_to_lds_b32) && \
    __has_builtin(__builtin_amdgcn_s_wait_asynccnt)
#define USE_ASYNC_LDS 1
#else
#define USE_ASYNC_LDS 0
#endif

typedef __attribute__((address_space(1))) int g_i32_t;   // global dword
typedef __attribute__((address_space(3))) int l_i32_t;   // LDS dword

// ---------------------------------------------------------------------------
// SpMM scatter:  dst[didx[e], :] += vals[e] * src[sidx[e], :]
// Block = 256 threads = 8 wave32s.  Edge metadata (sidx/didx/vals) for a
// 256-edge tile is staged into LDS with async global->LDS B32 copies
// (ASYNCcnt), double-buffered so tile t+1 streams in while tile t is
// processed.  Each wave then owns 32 edges of the tile: float4 row gather +
// 4 native global_atomic_add_f32 per lane, with global_prefetch_b8 on the
// next edge's source row (index already in LDS -> early prefetch address).
// ---------------------------------------------------------------------------
__global__ __launch_bounds__(256) void spmm_scatter(
    const float* __restrict__ src, float* __restrict__ dst,
    const int* __restrict__ sidx, const int* __restrict__ didx,
    const float* __restrict__ vals, int nnz)
{
    __shared__ int   s_si[2][TILE];
    __shared__ int   s_di[2][TILE];
    __shared__ float s_v [2][TILE];

    const int t    = threadIdx.x;          // 0..255: staging slot
    const int lane = threadIdx.x & 31;
    const int wv   = threadIdx.x >> 5;     // 0..7: wave in block
    const int col  = lane << 2;            // 4 floats per lane
    const int ntiles = (nnz + TILE - 1) / TILE;

    // --- stage tile `tl` metadata into LDS buffer `b` -----------------------
    auto stage = [&](int tl, int b) {
        const int idx = tl * TILE + t;
        if (idx < nnz) {
#if USE_ASYNC_LDS
            __builtin_amdgcn_global_load_async_to_lds_b32(
                (g_i32_t*)(sidx + idx), (l_i32_t*)&s_si[b][t], 0, 0);
            __builtin_amdgcn_global_load_async_to_lds_b32(
                (g_i32_t*)(didx + idx), (l_i32_t*)&s_di[b][t], 0, 0);
            __builtin_amdgcn_global_load_async_to_lds_b32(
                (g_i32_t*)(vals + idx), (l_i32_t*)&s_v[b][t], 0, 0);
#else
            s_si[b][t] = sidx[idx];
            s_di[b][t] = didx[idx];
            s_v [b][t] = vals[idx];
#endif
        }
    };

    int tile = blockIdx.x;
    if (tile < ntiles) stage(tile, 0);
    int buf = 0;

    for (; tile < ntiles; tile += gridDim.x, buf ^= 1) {
#if USE_ASYNC_LDS
        __builtin_amdgcn_s_wait_asynccnt(0);   // own async copies landed
#endif
        __syncthreads();                       // tile`buf` visible to all waves

        const int next = tile + gridDim.x;
        if (next < ntiles) stage(next, buf ^ 1);   // overlap next tile's stream

        // --- process 32 edges of this tile per wave -------------------------
        const int ebase = tile * TILE + (wv << 5);
#pragma unroll 1
        for (int j = 0; j < 32; ++j) {
            const int e = ebase + j;
            if (e >= nnz) break;
            const int slot = (wv << 5) + j;
            if (j + 1 < 32 && e + 1 < nnz) {
                // CDNA5 global_prefetch_b8: warm next gather row (index in LDS)
                __builtin_prefetch(
                    src + (size_t)s_si[buf][slot + 1] * DIM + col, 0, 1);
            }
            const int   si = s_si[buf][slot];
            const int   di = s_di[buf][slot];
            const float v  = s_v [buf][slot];
            const float4 x = *(const float4*)(src + (size_t)si * DIM + col);
            float* p = dst + (size_t)di * DIM + col;
            unsafeAtomicAdd(p + 0, v * x.x);   // global_atomic_add_f32 (L2)
            unsafeAtomicAdd(p + 1, v * x.y);
            unsafeAtomicAdd(p + 2, v * x.z);
            unsafeAtomicAdd(p + 3, v * x.w);
        }
        __syncthreads();   // all reads of `buf` done before it is restaged
    }
}

// ---------------------------------------------------------------------------
// Fused LeakyReLU (optional) + LayerNorm(gamma,beta) + residual(ego).
// One wave32 per 128-wide row; butterfly reduction across the 32 lanes.
// ---------------------------------------------------------------------------
__global__ __launch_bounds__(256) void ln_leaky_res(
    const float* __restrict__ h,     const float* __restrict__ gamma,
    const float* __restrict__ beta,  const float* __restrict__ res,
    float* __restrict__ out, int nrows, int do_leaky)
{
    const int lane = threadIdx.x & 31;
    const int col  = lane << 2;
    const int row  = blockIdx.x * (blockDim.x >> 5) + (threadIdx.x >> 5);
    if (row >= nrows) return;

    const size_t base = (size_t)row * DIM + col;
    float4 x = *(const float4*)(h + base);
    if (do_leaky) {
        x.x = x.x >= 0.f ? x.x : LEAKY_K * x.x;
        x.y = x.y >= 0.f ? x.y : LEAKY_K * x.y;
        x.z = x.z >= 0.f ? x.z : LEAKY_K * x.z;
        x.w = x.w >= 0.f ? x.w : LEAKY_K * x.w;
    }

    float s = x.x + x.y + x.z + x.w;
#pragma unroll
    for (int off = 16; off > 0; off >>= 1) s += __shfl_xor(s, off, 32);
    const float mu = s * (1.0f / DIM);

    float4 d;
    d.x = x.x - mu; d.y = x.y - mu; d.z = x.z - mu; d.w = x.w - mu;
    float q = d.x * d.x + d.y * d.y + d.z * d.z + d.w * d.w;
#pragma unroll
    for (int off = 16; off > 0; off >>= 1) q += __shfl_xor(q, off, 32);
    const float rstd = rsqrtf(q * (1.0f / DIM) + LN_EPS_K);

    const float4 g = *(const float4*)(gamma + col);
    const float4 b = *(const float4*)(beta  + col);
    const float4 r = *(const float4*)(res + base);
    float4 o;
    o.x = d.x * rstd * g.x + b.x + r.x;
    o.y = d.y * rstd * g.y + b.y + r.y;
    o.z = d.z * rstd * g.z + b.z + r.z;
    o.w = d.w * rstd * g.w + b.w + r.w;
    *(float4*)(out + base) = o;
}

// ---------------------------------------------------------------------------
// Driver.  d_in order: ego, adj_vals, ln_gamma, ln_beta, adj_rows, adj_cols,
// n_users.  Output = concat(user_emb, item_emb) = full final embedding matrix.
// Workspace: ht + h (2 x 25.6 MB).  Intermediate layer output reuses d_out.
// ---------------------------------------------------------------------------
extern "C" void kernel_launch(void* const* d_in, const int* in_sizes, int n_in,
                              void* d_out, int out_size, void* d_ws, size_t ws_size,
                              hipStream_t stream)
{
    const float* ego   = (const float*)d_in[0];
    const float* vals  = (const float*)d_in[1];
    const float* gamma = (const float*)d_in[2];
    const float* beta  = (const float*)d_in[3];
    const int*   rows  = (const int*)d_in[4];
    const int*   cols  = (const int*)d_in[5];

    const int nnz     = in_sizes[1];
    const int nnodes  = in_sizes[0] / DIM;
    const int nlayers = in_sizes[2] / DIM;

    float* out = (float*)d_out;
    float* ht  = (float*)d_ws;
    float* h   = ht + (size_t)nnodes * DIM;
    const size_t tbl_bytes = (size_t)nnodes * DIM * sizeof(float);

    const dim3 blk(256);
    const int ntiles      = (nnz + TILE - 1) / TILE;
    const int spmm_blocks = (ntiles < 1024) ? ntiles : 1024;  // ~3 tiles/block
    const int ln_blocks   = (nnodes + 7) / 8;                 // 8 rows per block

    const float* embs = ego;
    for (int k = 0; k < nlayers; ++k) {
        // ht[cols[e]] += vals[e] * embs[rows[e]]
        (void)hipMemsetAsync(ht, 0, tbl_bytes, stream);
        spmm_scatter<<<spmm_blocks, blk, 0, stream>>>(embs, ht, rows, cols, vals, nnz);
        // h[rows[e]] += vals[e] * ht[cols[e]]
        (void)hipMemsetAsync(h, 0, tbl_bytes, stream);
        spmm_scatter<<<spmm_blocks, blk, 0, stream>>>(ht, h, cols, rows, vals, nnz);
        // embs_{k+1} = LN(leaky?(h)) * gamma_k + beta_k + ego   (-> d_out)
        ln_leaky_res<<<ln_blocks, blk, 0, stream>>>(
            h, gamma + (size_t)k * DIM, beta + (size_t)k * DIM,
            ego, out, nnodes, (k != nlayers - 1) ? 1 : 0);
        embs = out;
    }
}